// ProjectAndSample_57930518888727
// MI455X (gfx1250) — compile-verified
//
#include <hip/hip_runtime.h>
#include <cstdint>
#include <cstddef>

// ---------------------------------------------------------------------------
// Problem constants (from the reference): B=2, V=7, C=32, H=W=S=64.
// ANGLES[v] = -90 + 30*v degrees; FOV=80mm -> half_fov=40; center=(0,30,0).
// ---------------------------------------------------------------------------
#define BB   2
#define VV   7
#define CC   32
#define SS   64

typedef __attribute__((ext_vector_type(4))) unsigned int v4u;
typedef __attribute__((ext_vector_type(8))) int          v8i;
typedef __attribute__((ext_vector_type(4))) int          v4i;

#if defined(__has_builtin)
# if __has_builtin(__builtin_amdgcn_tensor_load_to_lds)
#  define HAVE_TDM 1
# endif
#endif
#ifndef HAVE_TDM
# define HAVE_TDM 0
#endif

// ---- CDNA5 async LDS<->global helpers (ASYNCcnt-tracked, ISA §10 / §15.18) ----
__device__ __forceinline__ void async_load_b32_to_lds(uint32_t lds_byte_addr,
                                                      const void* gaddr) {
  asm volatile("global_load_async_to_lds_b32 %0, %1, off"
               :: "v"(lds_byte_addr), "v"(gaddr) : "memory");
}
__device__ __forceinline__ void async_store_b32_from_lds(void* gaddr,
                                                         uint32_t lds_byte_addr) {
  asm volatile("global_store_async_from_lds_b32 %0, %1, off"
               :: "v"(gaddr), "v"(lds_byte_addr) : "memory");
}
__device__ __forceinline__ void wait_asynccnt0() {
  asm volatile("s_wait_asynccnt 0" ::: "memory");
}
__device__ __forceinline__ void wait_tensorcnt0() {
#if defined(__has_builtin)
# if __has_builtin(__builtin_amdgcn_s_wait_tensorcnt)
  __builtin_amdgcn_s_wait_tensorcnt(0);
  return;
# endif
#endif
  asm volatile("s_wait_tensorcnt 0" ::: "memory");
}

// ---------------------------------------------------------------------------
// Kernel 1: transpose feat_maps [B,V,C,H,W] -> fmT [B,V,H,W,C] via LDS.
// One block per (bv, y) row: moves a [C=32 x W=64] slab (8 KB).
// LDS layout: row-major [c][x] with padded stride 65 dwords (bank-conflict free
// on the transposed read; matches TDM pad_interval=64dw / pad_amount=1dw).
// ---------------------------------------------------------------------------
__global__ void __launch_bounds__(256)
transpose_feat_kernel(const float* __restrict__ feat, float* __restrict__ fmT) {
  const int H = SS, W = SS, C = CC;
  const int blk = blockIdx.x;
  const int y   = blk & (H - 1);     // 0..63
  const int bv  = blk >> 6;          // 0..13

  __shared__ float smem[CC * 65];    // 32 rows * 65 dwords = 8320 B
  const uint32_t lds_base = (uint32_t)(uintptr_t)(&smem[0]);

  const float* src = feat + ((size_t)bv * C) * (H * W) + (size_t)y * W; // (c=0,x=0)
  const int t = threadIdx.x;

#if HAVE_TDM
  // -------- Phase A: Tensor Data Mover copies the slab into LDS ----------
  if (t < 32) {  // wave 0 only (TENSORcnt is per-wave)
    const uint64_t gaddr = (uint64_t)(uintptr_t)src;
    // D# group 0: count=1, lds_addr, global_addr[56:0], type=2 ("image")
    v4u g0 = { 1u,
               lds_base,
               (uint32_t)(gaddr & 0xFFFFFFFFu),
               (uint32_t)((gaddr >> 32) & 0x01FFFFFFu) | 0x80000000u };
    // D# group 1:
    //  w0: data_size=2 (4B) | pad_enable | pad_interval=5 (64 dw) | pad_amount=0 (1 dw)
    //  tensor_dim0=64, tensor_dim1=32, tile_dim0=64, tile_dim1=32,
    //  tensor_dim0_stride=4096 elements (H*W), dim1_stride unused (2D tile).
    v8i g1 = { (int)0x01520000,
               (int)(64u << 16),     // tensor_dim0[15:0] @bits63:48
               (int)(32u << 16),     // tensor_dim1[15:0] @bits95:80
               (int)(64u << 16),     // tile_dim0 @bits127:112
               (int)32,              // tile_dim1 @bits143:128
               (int)4096,            // tensor_dim0_stride low32
               0, 0 };
    v4i gz = { 0, 0, 0, 0 };
# if __clang_major__ >= 23
    v8i gz8 = { 0, 0, 0, 0, 0, 0, 0, 0 };
    __builtin_amdgcn_tensor_load_to_lds(g0, g1, gz, gz, gz8, 0);
# else
    __builtin_amdgcn_tensor_load_to_lds(g0, g1, gz, gz, 0);
# endif
    wait_tensorcnt0();
  }
#else
  // -------- Phase A fallback: per-lane async scatter into LDS ------------
  for (int i = 0; i < 8; ++i) {
    const int e = i * 256 + t;
    const int c = e >> 6;            // 0..31
    const int x = e & 63;            // 0..63 (consecutive lanes -> coalesced)
    async_load_b32_to_lds(lds_base + (uint32_t)(c * 65 + x) * 4u,
                          src + (size_t)c * (H * W) + x);
  }
  wait_asynccnt0();
#endif
  __syncthreads();

  // -------- Phase B: async DMA LDS -> global in transposed [x][c] order ---
  float* dst = fmT + ((size_t)bv * H * W) * C + (size_t)y * W * C;
  for (int i = 0; i < 8; ++i) {
    const int e = i * 256 + t;
    const int c = e & 31;            // consecutive lanes -> consecutive c (128B)
    const int x = e >> 5;            // 0..63
    async_store_b32_from_lds(dst + (size_t)x * C + c,
                             lds_base + (uint32_t)(c * 65 + x) * 4u);
  }
  wait_asynccnt0();
}

// ---------------------------------------------------------------------------
// Kernel 2: project + bilinear sample. One wave32 per (b,n,v); lane = channel.
// All 4 corner reads and the output store are single 128 B wave transactions
// (TRANSPOSED path). Feature stack (7.3 MB) stays L2-resident on MI455X
// (192 MB L2) because the 180 MB of write-once output goes out non-temporal.
// Invisible point-views (feat == 0 exactly in the reference) early-out and
// skip all gather traffic — the branch is wave-uniform.
// ---------------------------------------------------------------------------
template <bool TRANSPOSED>
__global__ void __launch_bounds__(256)
project_sample_kernel(const float* __restrict__ coords,
                      const float* __restrict__ fm,
                      float* __restrict__ out_feat,
                      float* __restrict__ out_vis,
                      int N, int PV) {
  const int H = SS, W = SS, C = CC;
  const int wid  = blockIdx.x * (blockDim.x >> 5) + (threadIdx.x >> 5);
  const int lane = threadIdx.x & 31;
  if (wid >= PV) return;

  const int v  = wid % VV;
  const int bn = wid / VV;           // b*N + n
  const int b  = bn / N;

  // coords are streamed exactly once -> non-temporal loads
  const float xw = __builtin_nontemporal_load(&coords[(size_t)bn * 3 + 0]) - 0.0f;
  const float yw = __builtin_nontemporal_load(&coords[(size_t)bn * 3 + 1]) - 30.0f;
  const float zw = __builtin_nontemporal_load(&coords[(size_t)bn * 3 + 2]) - 0.0f;

  const float ang = (30.0f * (float)v - 90.0f) * 0.017453292519943295f;
  const float ca = cosf(ang);
  const float sa = sinf(ang);

  const float inv_half = 1.0f / 40.0f;
  const float u  = (xw * ca + zw * sa) * inv_half;
  const float vn = yw * inv_half;
  const bool vis = (fabsf(u) <= 1.0f) && (fabsf(vn) <= 1.0f);

  float* feat_ptr = &out_feat[(size_t)wid * C + lane];

  if (!vis) {                                   // wave-uniform early-out
    __builtin_nontemporal_store(0.0f, feat_ptr);
    if (lane == 0) __builtin_nontemporal_store(0.0f, &out_vis[wid]);
    return;
  }

  // grid_sample bilinear, zeros padding, align_corners=False
  const float ix = ((u  + 1.0f) * (float)W - 1.0f) * 0.5f;
  const float iy = ((vn + 1.0f) * (float)H - 1.0f) * 0.5f;
  const float x0f = floorf(ix), y0f = floorf(iy);
  const float wx1 = ix - x0f, wx0 = 1.0f - wx1;
  const float wy1 = iy - y0f, wy0 = 1.0f - wy1;
  const int x0 = (int)x0f, y0 = (int)y0f;
  const int bv = b * VV + v;

  float acc = 0.0f;
  #pragma unroll
  for (int k = 0; k < 4; ++k) {
    const int yi = y0 + (k >> 1);
    const int xi = x0 + (k & 1);
    const float w = (k == 0 ? wy0 * wx0 : k == 1 ? wy0 * wx1
                   : k == 2 ? wy1 * wx0 : wy1 * wx1);
    if (yi >= 0 && yi < H && xi >= 0 && xi < W) {   // wave-uniform branch
      size_t idx;
      if (TRANSPOSED)
        idx = (((size_t)bv * H + yi) * W + xi) * C + lane;     // 128B coalesced
      else
        idx = (((size_t)bv * C + lane) * H + yi) * W + xi;     // strided fallback
      acc += w * fm[idx];
    }
  }

  __builtin_nontemporal_store(acc, feat_ptr);     // 128B coalesced, TH=NT
  if (lane == 0) __builtin_nontemporal_store(1.0f, &out_vis[wid]);
}

// ---------------------------------------------------------------------------
extern "C" void kernel_launch(void* const* d_in, const int* in_sizes, int n_in,
                              void* d_out, int out_size, void* d_ws, size_t ws_size,
                              hipStream_t stream) {
  (void)n_in; (void)out_size;
  const float* coords = (const float*)d_in[0];   // [B,N,3] f32
  const float* feat   = (const float*)d_in[1];   // [B,V,C,S,S] f32

  const int N  = in_sizes[0] / (BB * 3);
  const int PV = BB * N * VV;

  float* out_feat = (float*)d_out;                       // [B,N,V,C]
  float* out_vis  = out_feat + (size_t)PV * CC;          // [B,N,V] as 0/1 f32

  const size_t fmT_bytes = (size_t)BB * VV * CC * SS * SS * sizeof(float);
  const int sample_blocks = (PV + 7) / 8;                // 8 waves/block

  if (ws_size >= fmT_bytes) {
    float* fmT = (float*)d_ws;
    transpose_feat_kernel<<<BB * VV * SS, 256, 0, stream>>>(feat, fmT);
    project_sample_kernel<true><<<sample_blocks, 256, 0, stream>>>(
        coords, fmT, out_feat, out_vis, N, PV);
  } else {
    project_sample_kernel<false><<<sample_blocks, 256, 0, stream>>>(
        coords, feat, out_feat, out_vis, N, PV);
  }
}